// Jansen_Layer_83124797047476
// MI455X (gfx1250) — compile-verified
//
#include <hip/hip_runtime.h>
#include <hip/hip_bf16.h>
#include <stdint.h>

// Jansen-Rit Euler integration, one region per lane, one wave per block.
// History drained via gfx1250 async LDS->global stores (ASYNCcnt), 4-deep ring.

#define LANES 32
#define NBUF  4   // LDS ring depth (power of two)

__device__ __forceinline__ float fast_exp2(float x) {
#if __has_builtin(__builtin_amdgcn_exp2f)
  return __builtin_amdgcn_exp2f(x);     // v_exp_f32
#else
  return exp2f(x);
#endif
}

__device__ __forceinline__ float fast_rcp(float x) {
#if __has_builtin(__builtin_amdgcn_rcpf)
  return __builtin_amdgcn_rcpf(x);      // v_rcp_f32
#else
  return 1.0f / x;
#endif
}

__global__ __launch_bounds__(LANES) void jansen_rit_kernel(
    const float* __restrict__ init_state,
    const float* __restrict__ p_dt,
    const int*   __restrict__ p_nsteps,
    const float* __restrict__ pA,  const float* __restrict__ pa,
    const float* __restrict__ pB,  const float* __restrict__ pb,
    const float* __restrict__ pc1, const float* __restrict__ pc2,
    const float* __restrict__ pc3, const float* __restrict__ pc4,
    const float* __restrict__ pvmax, const float* __restrict__ pv0,
    const float* __restrict__ pr,  const float* __restrict__ pstd,
    float* __restrict__ out, int R)
{
  __shared__ __align__(16) float lbuf[NBUF * LANES * 6];

  const int lane   = threadIdx.x;
  const int region = blockIdx.x * LANES + lane;
  const bool active = region < R;
  const int  nreg   = min(LANES, R - blockIdx.x * LANES);
  const unsigned chunkBytes = (unsigned)nreg * 24u;   // bytes this block emits per step

  // ---- scalar parameters (uniform loads) ----
  const float dt     = p_dt[0];
  const int   nsteps = p_nsteps[0];
  const float A = pA[0],  a = pa[0],  B = pB[0],  b = pb[0];
  const float c1 = pc1[0], c2 = pc2[0], c3 = pc3[0], c4 = pc4[0];
  const float vmax = pvmax[0], v0 = pv0[0], rr = pr[0], std_in = pstd[0];

  // ---- hoisted constants: sig(x) = vmax * rcp(1 + exp2(rl2*v0 - rl2*x)) ----
  const float rl2    = rr * 1.4426950408889634f;  // r * log2(e)
  const float rl2v0  = rl2 * v0;
  const float rl2c1  = rl2 * c1;
  const float rl2c3  = rl2 * c3;
  const float kA     = A * a;
  const float kAvm   = kA * vmax;        // A*a*sig(E-I) term
  const float kAc2vm = kA * c2 * vmax;   // A*a*c2*sig(c1*M) term
  const float kAstd  = kA * std_in;
  const float kBc4vm = B * b * c4 * vmax;
  const float a2 = a * a, twoa = 2.0f * a;
  const float b2 = b * b, twob = 2.0f * b;

  // ---- per-lane state in registers ----
  float M = 0.f, E = 0.f, I = 0.f, Mv = 0.f, Ev = 0.f, Iv = 0.f;
  if (active) {
    const float* s = init_state + (size_t)region * 6;
    M = s[0]; E = s[1]; I = s[2]; Mv = s[3]; Ev = s[4]; Iv = s[5];
  }

  // ---- addresses for the async LDS->global drain ----
  // d_out layout: [R,6] final state, then [steps,R,6] history
  float* hist = out + (size_t)R * 6;
  unsigned long long gaddr =
      (unsigned long long)(uintptr_t)hist +
      (unsigned long long)blockIdx.x * (LANES * 24u) +
      (unsigned long long)lane * 8u;
  const unsigned long long rowStride = (unsigned long long)R * 24u;

  // low 32 bits of a flat pointer into LDS == LDS byte offset (aperture in [63:32])
  const unsigned lbase = (unsigned)(uintptr_t)&lbuf[0];

  for (int t = 0; t < nsteps; ++t) {
    // ---- derivatives from OLD state (matches reference ordering) ----
    const float eEI = fast_exp2(fmaf(-rl2,   E - I, rl2v0));
    const float eM1 = fast_exp2(fmaf(-rl2c1, M,     rl2v0));
    const float eM3 = fast_exp2(fmaf(-rl2c3, M,     rl2v0));
    const float dMv = fmaf(kAvm,   fast_rcp(1.0f + eEI), fmaf(-twoa, Mv, -a2 * M));
    const float dEv = fmaf(kAc2vm, fast_rcp(1.0f + eM1),
                           fmaf(-twoa, Ev, fmaf(-a2, E, kAstd)));
    const float dIv = fmaf(kBc4vm, fast_rcp(1.0f + eM3), fmaf(-twob, Iv, -b2 * I));

    // ---- Euler update (positions use OLD velocities) ----
    M  = fmaf(dt, Mv,  M);
    E  = fmaf(dt, Ev,  E);
    I  = fmaf(dt, Iv,  I);
    Mv = fmaf(dt, dMv, Mv);
    Ev = fmaf(dt, dEv, Ev);
    Iv = fmaf(dt, dIv, Iv);

    // ---- ring-slot reuse guard: stores issued NBUF steps ago must be done ----
    asm volatile("s_wait_asynccnt 0x9" ::: "memory");   // <= 3*(NBUF-1) outstanding

    // ---- stage this step's 24B/lane into LDS (bank-conflict-free) ----
    float2* bp = (float2*)&lbuf[(t & (NBUF - 1)) * (LANES * 6) + lane * 6];
    bp[0] = make_float2(M,  E);
    bp[1] = make_float2(I,  Mv);
    bp[2] = make_float2(Ev, Iv);

    // LDS data must be visible to the async engine before it reads
    asm volatile("s_wait_dscnt 0x0" ::: "memory");

    // ---- async LDS->global: 3 x b64 per lane-slot, IOFFSET shifts both sides ----
    const unsigned la = lbase + (unsigned)(t & (NBUF - 1)) * (LANES * 24u)
                              + (unsigned)lane * 8u;
    const unsigned lb = (unsigned)lane * 8u;
    if (lb < chunkBytes)
      asm volatile("global_store_async_from_lds_b64 %0, %1, off"
                   :: "v"(gaddr), "v"(la) : "memory");
    if (lb + 256u < chunkBytes)
      asm volatile("global_store_async_from_lds_b64 %0, %1, off offset:256"
                   :: "v"(gaddr), "v"(la) : "memory");
    if (lb + 512u < chunkBytes)
      asm volatile("global_store_async_from_lds_b64 %0, %1, off offset:512"
                   :: "v"(gaddr), "v"(la) : "memory");

    gaddr += rowStride;
  }

  // ---- final state block [R,6] at the front of d_out ----
  if (active) {
    float2* o = (float2*)(out + (size_t)region * 6);
    o[0] = make_float2(M,  E);
    o[1] = make_float2(I,  Mv);
    o[2] = make_float2(Ev, Iv);
  }

  // all async traffic drained by implicit wait-idle at s_endpgm
}

extern "C" void kernel_launch(void* const* d_in, const int* in_sizes, int n_in,
                              void* d_out, int out_size, void* d_ws, size_t ws_size,
                              hipStream_t stream) {
  (void)n_in; (void)out_size; (void)d_ws; (void)ws_size;
  const float* init_state = (const float*)d_in[0];
  const float* step_size  = (const float*)d_in[1];
  const int*   num_steps  = (const int*)  d_in[2];
  const float* A    = (const float*)d_in[3];
  const float* a    = (const float*)d_in[4];
  const float* B    = (const float*)d_in[5];
  const float* b    = (const float*)d_in[6];
  const float* c1   = (const float*)d_in[7];
  const float* c2   = (const float*)d_in[8];
  const float* c3   = (const float*)d_in[9];
  const float* c4   = (const float*)d_in[10];
  const float* vmax = (const float*)d_in[11];
  const float* v0   = (const float*)d_in[12];
  const float* r    = (const float*)d_in[13];
  const float* std_ = (const float*)d_in[14];

  const int R = in_sizes[0] / 6;           // 200 regions
  const int blocks = (R + LANES - 1) / LANES;  // 7 single-wave blocks

  hipLaunchKernelGGL(jansen_rit_kernel, dim3(blocks), dim3(LANES), 0, stream,
                     init_state, step_size, num_steps,
                     A, a, B, b, c1, c2, c3, c4, vmax, v0, r, std_,
                     (float*)d_out, R);
}